// Kernel_3792100198503982614_60842506715273
// MI455X (gfx1250) — compile-verified
//
#include <hip/hip_runtime.h>

typedef __attribute__((ext_vector_type(2))) float v2f;
typedef __attribute__((ext_vector_type(8))) float v8f;

#define NB   32
#define CC   256
#define HWD  3136
#define KG   128    // C / G
#define LDSP 18     // LDS row pitch in floats: 8B-aligned, bank-conflict-free strides

#define WMMA_F32(acc, a, b) \
  acc = __builtin_amdgcn_wmma_f32_16x16x4_f32(false, (a), false, (b), (short)0, (acc), false, false)

// ---------------------------------------------------------------------------
// GEMM 1: t2[n] = (p1 .* x[n]) [256 x 3136]  @  p2 [3136 x 128]
// A (p1*x) staged through LDS with coalesced float4 loads; B (p2) is
// lane-contiguous in global, read directly.
// ---------------------------------------------------------------------------
__global__ __launch_bounds__(256) void k_gemm_t2(const float* __restrict__ x,
                                                 const float* __restrict__ p1,
                                                 const float* __restrict__ p2,
                                                 float* __restrict__ t2w) {
  __shared__ float As[64][LDSP];
  const int tid = threadIdx.x;
  const int lane = tid & 31, wave = tid >> 5;
  const int wm = wave >> 1, wn = wave & 1;
  const int n = blockIdx.z;
  const int bm0 = blockIdx.y * 64;
  const int n0 = blockIdx.x * 128 + wn * 64;
  const int half = lane >> 4, l15 = lane & 15;
  const int srow = tid >> 2, scol = (tid & 3) * 4;     // staging: 4 thr/row, 16B each

  const float* xr = x + ((size_t)n * CC + bm0 + srow) * HWD + scol;
  const float* pr = p1 + (size_t)(bm0 + srow) * HWD + scol;

  v8f acc0 = {}, acc1 = {}, acc2 = {}, acc3 = {};
  for (int k0 = 0; k0 < HWD; k0 += 16) {
    __syncthreads();
    {
      float4 xv = *(const float4*)(xr + k0);
      float4 pv = *(const float4*)(pr + k0);
      As[srow][scol + 0] = xv.x * pv.x;
      As[srow][scol + 1] = xv.y * pv.y;
      As[srow][scol + 2] = xv.z * pv.z;
      As[srow][scol + 3] = xv.w * pv.w;
    }
    __syncthreads();
#pragma unroll
    for (int kk = 0; kk < 16; kk += 4) {
      const int ka = kk + 2 * half;
      v2f a = *(const v2f*)&As[wm * 16 + l15][ka];
      const float* bp = p2 + (size_t)(k0 + ka) * KG + n0 + l15;
      v2f b0, b1, b2, b3;
      b0.x = bp[0];  b0.y = bp[KG];
      b1.x = bp[16]; b1.y = bp[KG + 16];
      b2.x = bp[32]; b2.y = bp[KG + 32];
      b3.x = bp[48]; b3.y = bp[KG + 48];
      WMMA_F32(acc0, a, b0);
      WMMA_F32(acc1, a, b1);
      WMMA_F32(acc2, a, b2);
      WMMA_F32(acc3, a, b3);
    }
  }
  float* out = t2w + ((size_t)n * CC + bm0 + wm * 16 + 8 * half) * KG + n0 + l15;
#pragma unroll
  for (int v = 0; v < 8; ++v) {
    out[(size_t)v * KG + 0]  = acc0[v];
    out[(size_t)v * KG + 16] = acc1[v];
    out[(size_t)v * KG + 32] = acc2[v];
    out[(size_t)v * KG + 48] = acc3[v];
  }
}

// ---------------------------------------------------------------------------
// t4: grouped dilated conv on t2 (only kw==1 survives; width==1, pad 3, dil 3)
// ---------------------------------------------------------------------------
__global__ __launch_bounds__(256) void k_conv_t4(const float* __restrict__ t2w,
                                                 const float* __restrict__ w4,
                                                 float* __restrict__ t4w) {
  unsigned idx = blockIdx.x * 256u + threadIdx.x;   // exactly 32*256*128 threads
  unsigned y = idx & 127u;
  unsigned co = (idx >> 7) & 255u;
  unsigned n = idx >> 15;
  unsigned g = co >> 2;
  const float* tp = t2w + ((size_t)n * CC + g * 4u) * KG;
  const float* wp = w4 + (size_t)co * 36u;
  float s = 0.f;
#pragma unroll
  for (int kh = 0; kh < 3; ++kh) {
    int yy = (int)y + 3 * (kh - 1);
    if (yy < 0 || yy > 127) continue;
#pragma unroll
    for (int ci = 0; ci < 4; ++ci)
      s += wp[ci * 9 + kh * 3 + 1] * tp[ci * KG + yy];
  }
  t4w[idx] = s;
}

// ---------------------------------------------------------------------------
// GEMM 2: t3[n] = (p1 .* x[n]) [256x3136] @ x[n]^T [3136x256]  * (1/56)
// Both operands are lane-strided in global -> both staged through LDS.
// ---------------------------------------------------------------------------
__global__ __launch_bounds__(256) void k_gemm_t3(const float* __restrict__ x,
                                                 const float* __restrict__ p1,
                                                 float* __restrict__ t3w) {
  __shared__ float As[64][LDSP];
  __shared__ float Bs[128][LDSP];
  const int tid = threadIdx.x;
  const int lane = tid & 31, wave = tid >> 5;
  const int wm = wave >> 1, wn = wave & 1;
  const int n = blockIdx.z;
  const int bm0 = blockIdx.y * 64;
  const int bn0 = blockIdx.x * 128;
  const int half = lane >> 4, l15 = lane & 15;
  const int srow = tid >> 2, scol = (tid & 3) * 4;     // A staging: 4 thr/row
  const int brow = tid >> 1, bcol = (tid & 1) * 8;     // B staging: 2 thr/row, 32B each

  const float* xr = x + ((size_t)n * CC + bm0 + srow) * HWD + scol;
  const float* pr = p1 + (size_t)(bm0 + srow) * HWD + scol;
  const float* br = x + ((size_t)n * CC + bn0 + brow) * HWD + bcol;

  v8f acc0 = {}, acc1 = {}, acc2 = {}, acc3 = {};
  for (int k0 = 0; k0 < HWD; k0 += 16) {
    __syncthreads();
    {
      float4 xv = *(const float4*)(xr + k0);
      float4 pv = *(const float4*)(pr + k0);
      As[srow][scol + 0] = xv.x * pv.x;
      As[srow][scol + 1] = xv.y * pv.y;
      As[srow][scol + 2] = xv.z * pv.z;
      As[srow][scol + 3] = xv.w * pv.w;
      float4 b0 = *(const float4*)(br + k0);
      float4 b1 = *(const float4*)(br + k0 + 4);
      Bs[brow][bcol + 0] = b0.x; Bs[brow][bcol + 1] = b0.y;
      Bs[brow][bcol + 2] = b0.z; Bs[brow][bcol + 3] = b0.w;
      Bs[brow][bcol + 4] = b1.x; Bs[brow][bcol + 5] = b1.y;
      Bs[brow][bcol + 6] = b1.z; Bs[brow][bcol + 7] = b1.w;
    }
    __syncthreads();
#pragma unroll
    for (int kk = 0; kk < 16; kk += 4) {
      const int ka = kk + 2 * half;
      v2f a = *(const v2f*)&As[wm * 16 + l15][ka];
      v2f b0 = *(const v2f*)&Bs[wn * 64 + 0  + l15][ka];
      v2f b1 = *(const v2f*)&Bs[wn * 64 + 16 + l15][ka];
      v2f b2 = *(const v2f*)&Bs[wn * 64 + 32 + l15][ka];
      v2f b3 = *(const v2f*)&Bs[wn * 64 + 48 + l15][ka];
      WMMA_F32(acc0, a, b0);
      WMMA_F32(acc1, a, b1);
      WMMA_F32(acc2, a, b2);
      WMMA_F32(acc3, a, b3);
    }
  }
  const float scale = 1.0f / 56.0f;   // 1/sqrt(3136)
  const int n0 = bn0 + wn * 64;
  float* out = t3w + ((size_t)n * CC + bm0 + wm * 16 + 8 * half) * CC + n0 + l15;
#pragma unroll
  for (int v = 0; v < 8; ++v) {
    out[(size_t)v * CC + 0]  = acc0[v] * scale;
    out[(size_t)v * CC + 16] = acc1[v] * scale;
    out[(size_t)v * CC + 32] = acc2[v] * scale;
    out[(size_t)v * CC + 48] = acc3[v] * scale;
  }
}

// ---------------------------------------------------------------------------
// GEMM 3: t5[n] = w5 [256x256] @ x[n] [256x3136]  -> t9 scratch
// ---------------------------------------------------------------------------
__global__ __launch_bounds__(256) void k_gemm_t5(const float* __restrict__ x,
                                                 const float* __restrict__ w5,
                                                 float* __restrict__ t9w) {
  __shared__ float As[64][LDSP];
  const int tid = threadIdx.x;
  const int lane = tid & 31, wave = tid >> 5;
  const int wm = wave >> 1, wn = wave & 1;
  const int n = blockIdx.z;
  const int bm0 = blockIdx.y * 64;
  const int n0 = blockIdx.x * 128 + wn * 64;
  const bool nvalid = (n0 < HWD);                  // 3136 % 64 == 0 -> wave-uniform
  const int n0c = nvalid ? n0 : 0;                 // no early return: staging + EXEC
  const int half = lane >> 4, l15 = lane & 15;
  const int srow = tid >> 2, scol = (tid & 3) * 4;

  const float* wr = w5 + (size_t)(bm0 + srow) * CC + scol;
  const float* xb = x + (size_t)n * CC * HWD;

  v8f acc0 = {}, acc1 = {}, acc2 = {}, acc3 = {};
  for (int k0 = 0; k0 < CC; k0 += 16) {
    __syncthreads();
    {
      float4 wv = *(const float4*)(wr + k0);
      As[srow][scol + 0] = wv.x;
      As[srow][scol + 1] = wv.y;
      As[srow][scol + 2] = wv.z;
      As[srow][scol + 3] = wv.w;
    }
    __syncthreads();
#pragma unroll
    for (int kk = 0; kk < 16; kk += 4) {
      const int ka = kk + 2 * half;
      v2f a = *(const v2f*)&As[wm * 16 + l15][ka];
      const float* bp = xb + (size_t)(k0 + ka) * HWD + n0c + l15;
      v2f b0, b1, b2, b3;
      b0.x = bp[0];  b0.y = bp[HWD];
      b1.x = bp[16]; b1.y = bp[HWD + 16];
      b2.x = bp[32]; b2.y = bp[HWD + 32];
      b3.x = bp[48]; b3.y = bp[HWD + 48];
      WMMA_F32(acc0, a, b0);
      WMMA_F32(acc1, a, b1);
      WMMA_F32(acc2, a, b2);
      WMMA_F32(acc3, a, b3);
    }
  }
  if (nvalid) {
    float* out = t9w + ((size_t)n * CC + bm0 + wm * 16 + 8 * half) * HWD + n0 + l15;
#pragma unroll
    for (int v = 0; v < 8; ++v) {
      out[(size_t)v * HWD + 0]  = acc0[v];
      out[(size_t)v * HWD + 16] = acc1[v];
      out[(size_t)v * HWD + 32] = acc2[v];
      out[(size_t)v * HWD + 48] = acc3[v];
    }
  }
}

// ---------------------------------------------------------------------------
// t9 = max(t5, t8) ; t8 = grouped (4-ch) dilated 3x3 conv on x
// ---------------------------------------------------------------------------
__global__ __launch_bounds__(256) void k_t8_max(const float* __restrict__ x,
                                                const float* __restrict__ w8,
                                                float* __restrict__ t9w) {
  unsigned idx = blockIdx.x * 256u + threadIdx.x;   // exactly 32*256*3136 threads
  unsigned i = idx % (unsigned)HWD;
  unsigned nc = idx / (unsigned)HWD;
  unsigned co = nc & 255u;
  unsigned n = nc >> 8;
  int h = (int)(i / 56u), w = (int)(i % 56u);
  unsigned g = co >> 2;
  const float* xb = x + ((size_t)n * CC + g * 4u) * HWD;
  const float* wp = w8 + (size_t)co * 36u;
  float s = 0.f;
#pragma unroll
  for (int kh = 0; kh < 3; ++kh) {
    int hh = h + 3 * (kh - 1);
    if (hh < 0 || hh > 55) continue;
#pragma unroll
    for (int kw = 0; kw < 3; ++kw) {
      int ww = w + 3 * (kw - 1);
      if (ww < 0 || ww > 55) continue;
#pragma unroll
      for (int ci = 0; ci < 4; ++ci)
        s += wp[ci * 9 + kh * 3 + kw] * xb[(size_t)ci * HWD + hh * 56 + ww];
    }
  }
  t9w[idx] = fmaxf(t9w[idx], s);
}

// ---------------------------------------------------------------------------
// GEMM 4: out[n] = t7[n] [256x256] @ t9[n] [256x3136] * (1/16)
// t7[n,c,d] = t4[n,c,d&127] - t3[n,c,d], built during LDS staging.
// ---------------------------------------------------------------------------
__global__ __launch_bounds__(256) void k_gemm_out(const float* __restrict__ t3w,
                                                  const float* __restrict__ t4w,
                                                  const float* __restrict__ t9w,
                                                  float* __restrict__ out) {
  __shared__ float As[64][LDSP];
  const int tid = threadIdx.x;
  const int lane = tid & 31, wave = tid >> 5;
  const int wm = wave >> 1, wn = wave & 1;
  const int n = blockIdx.z;
  const int bm0 = blockIdx.y * 64;
  const int n0 = blockIdx.x * 128 + wn * 64;
  const bool nvalid = (n0 < HWD);
  const int n0c = nvalid ? n0 : 0;
  const int half = lane >> 4, l15 = lane & 15;
  const int srow = tid >> 2, scol = (tid & 3) * 4;

  const float* a3 = t3w + ((size_t)n * CC + bm0 + srow) * CC + scol;
  const float* a4 = t4w + ((size_t)n * CC + bm0 + srow) * KG;
  const float* tb = t9w + (size_t)n * CC * HWD;

  v8f acc0 = {}, acc1 = {}, acc2 = {}, acc3 = {};
  for (int k0 = 0; k0 < CC; k0 += 16) {
    __syncthreads();
    {
      float4 v3 = *(const float4*)(a3 + k0);
      float4 v4 = *(const float4*)(a4 + ((k0 + scol) & (KG - 1)));
      As[srow][scol + 0] = v4.x - v3.x;
      As[srow][scol + 1] = v4.y - v3.y;
      As[srow][scol + 2] = v4.z - v3.z;
      As[srow][scol + 3] = v4.w - v3.w;
    }
    __syncthreads();
#pragma unroll
    for (int kk = 0; kk < 16; kk += 4) {
      const int ka = kk + 2 * half;
      v2f a = *(const v2f*)&As[wm * 16 + l15][ka];
      const float* bp = tb + (size_t)(k0 + ka) * HWD + n0c + l15;
      v2f b0, b1, b2, b3;
      b0.x = bp[0];  b0.y = bp[HWD];
      b1.x = bp[16]; b1.y = bp[HWD + 16];
      b2.x = bp[32]; b2.y = bp[HWD + 32];
      b3.x = bp[48]; b3.y = bp[HWD + 48];
      WMMA_F32(acc0, a, b0);
      WMMA_F32(acc1, a, b1);
      WMMA_F32(acc2, a, b2);
      WMMA_F32(acc3, a, b3);
    }
  }
  if (nvalid) {
    const float scale = 1.0f / 16.0f;   // 1/sqrt(256)
    float* op = out + ((size_t)n * CC + bm0 + wm * 16 + 8 * half) * HWD + n0 + l15;
#pragma unroll
    for (int v = 0; v < 8; ++v) {
      op[(size_t)v * HWD + 0]  = acc0[v] * scale;
      op[(size_t)v * HWD + 16] = acc1[v] * scale;
      op[(size_t)v * HWD + 32] = acc2[v] * scale;
      op[(size_t)v * HWD + 48] = acc3[v] * scale;
    }
  }
}

// ---------------------------------------------------------------------------
extern "C" void kernel_launch(void* const* d_in, const int* in_sizes, int n_in,
                              void* d_out, int out_size, void* d_ws, size_t ws_size,
                              hipStream_t stream) {
  const float* x  = (const float*)d_in[0];   // [32,256,56,56]
  const float* p1 = (const float*)d_in[1];   // [1,256,56,56] -> [256,3136]
  const float* p2 = (const float*)d_in[2];   // [56,56,128,1] -> [3136,128]
  const float* w4 = (const float*)d_in[3];   // [256,4,3,3]
  const float* w5 = (const float*)d_in[4];   // [256,256]
  const float* w8 = (const float*)d_in[5];   // [256,4,3,3]
  float* out = (float*)d_out;

  char* ws = (char*)d_ws;
  float* t2w = (float*)(ws);                                   // 4 MiB
  float* t4w = (float*)(ws + (size_t)NB * CC * KG * 4);        // 4 MiB
  float* t3w = (float*)(ws + (size_t)NB * CC * KG * 8);        // 8 MiB
  float* t9w = (float*)(ws + (size_t)NB * CC * KG * 8 +
                        (size_t)NB * CC * CC * 4);             // 98 MiB

  dim3 blk(256);

  k_gemm_t2<<<dim3(1, 4, NB), blk, 0, stream>>>(x, p1, p2, t2w);
  k_conv_t4<<<dim3((NB * CC * KG) / 256), blk, 0, stream>>>(t2w, w4, t4w);
  k_gemm_t3<<<dim3(2, 4, NB), blk, 0, stream>>>(x, p1, t3w);
  k_gemm_t5<<<dim3(25, 4, NB), blk, 0, stream>>>(x, w5, t9w);
  k_t8_max<<<dim3((NB * CC * HWD) / 256), blk, 0, stream>>>(x, w8, t9w);
  k_gemm_out<<<dim3(25, 4, NB), blk, 0, stream>>>(t3w, t4w, t9w, out);
}